// MyHGNN_25933012533354
// MI455X (gfx1250) — compile-verified
//
#include <hip/hip_runtime.h>

#define D 256

typedef __attribute__((ext_vector_type(2))) float v2f;
typedef __attribute__((ext_vector_type(8))) float v8f;

// ---------------------------------------------------------------------------
// Zero fill (float4 granularity; all sizes are multiples of 256 floats)
// ---------------------------------------------------------------------------
__global__ void hgnn_zero_kernel(float4* __restrict__ p, int n4) {
    int i = blockIdx.x * blockDim.x + threadIdx.x;
    if (i < n4) p[i] = make_float4(0.f, 0.f, 0.f, 0.f);
}

// ---------------------------------------------------------------------------
// H[M x 256] = relu(A[M x 256] @ W[256 x 256] + bias)
// One wave32 computes a 16(M) x 64(N) tile via 4 accumulators of
// V_WMMA_F32_16X16X4_F32 (the f32 WMMA shape on CDNA5).
//
// A-frag layout (ISA 7.12.2, 32-bit A 16x4): lanes 0-15 hold K=k,k+1,
// lanes 16-31 hold K=k+2,k+3 for row M=lane%16.
// B-frag (4x16, row striped across lanes): lane l holds col=l of rows
// K=k+2*half (vgpr0) and K=k+2*half+1 (vgpr1).
// D layout: VGPR r -> M = 16*tile + r + 8*half, N = lane%16 (+16 subtile).
// ---------------------------------------------------------------------------
__global__ void hgnn_gemm_bias_relu_wmma(const float* __restrict__ A,
                                         const float* __restrict__ W,
                                         const float* __restrict__ bias,
                                         float* __restrict__ H, int M) {
    int wave  = (blockIdx.x * blockDim.x + threadIdx.x) >> 5;
    int lane  = threadIdx.x & 31;
    int mtile = wave >> 2;   // 4 waves (n64 = 0..3) per M tile
    int n64   = wave & 3;
    if (mtile * 16 >= M) return;           // wave-uniform exit (EXEC stays full)

    int l    = lane & 15;
    int half = lane >> 4;
    int row  = mtile * 16 + l;
    if (row >= M) row = M - 1;             // clamp; masked at store
    const float* Ar = A + (long long)row * D;
    int col0 = n64 * 64 + l;

    v8f acc0 = {}, acc1 = {}, acc2 = {}, acc3 = {};

#pragma unroll 4
    for (int k = 0; k < D; k += 4) {
        int kk = k + 2 * half;
        v2f a;
        a.x = Ar[kk];
        a.y = Ar[kk + 1];
        const float* W0 = W + kk * D;
        const float* W1 = W0 + D;
        v2f b0, b1, b2, b3;
        b0.x = W0[col0 +  0]; b0.y = W1[col0 +  0];
        b1.x = W0[col0 + 16]; b1.y = W1[col0 + 16];
        b2.x = W0[col0 + 32]; b2.y = W1[col0 + 32];
        b3.x = W0[col0 + 48]; b3.y = W1[col0 + 48];
        acc0 = __builtin_amdgcn_wmma_f32_16x16x4_f32(false, a, false, b0, (short)0, acc0, false, false);
        acc1 = __builtin_amdgcn_wmma_f32_16x16x4_f32(false, a, false, b1, (short)0, acc1, false, false);
        acc2 = __builtin_amdgcn_wmma_f32_16x16x4_f32(false, a, false, b2, (short)0, acc2, false, false);
        acc3 = __builtin_amdgcn_wmma_f32_16x16x4_f32(false, a, false, b3, (short)0, acc3, false, false);
    }

    int cb = n64 * 64 + l;
    float bb0 = bias[cb +  0];
    float bb1 = bias[cb + 16];
    float bb2 = bias[cb + 32];
    float bb3 = bias[cb + 48];
#pragma unroll
    for (int r = 0; r < 8; ++r) {
        int orow = mtile * 16 + r + 8 * half;
        if (orow < M) {
            float* o = H + (long long)orow * D + cb;
            o[ 0] = fmaxf(acc0[r] + bb0, 0.f);
            o[16] = fmaxf(acc1[r] + bb1, 0.f);
            o[32] = fmaxf(acc2[r] + bb2, 0.f);
            o[48] = fmaxf(acc3[r] + bb3, 0.f);
        }
    }
}

// ---------------------------------------------------------------------------
// out[dst[e]] += H[src[e]] * w[e]   (one wave per edge, 8 floats per lane)
// ei is [2, E] row-major: ei[e] = src, ei[E + e] = dst.
// ---------------------------------------------------------------------------
__global__ void hgnn_scatter_edges(const float* __restrict__ H,
                                   const int* __restrict__ ei,
                                   const float* __restrict__ ew,
                                   float* __restrict__ out, int E_) {
    int gid = blockIdx.x * blockDim.x + threadIdx.x;
    int e = gid >> 5;
    if (e >= E_) return;
    int c   = gid & 31;
    int src = ei[e];
    int dst = ei[E_ + e];
    float w = ew[e];

    const float4* hs = (const float4*)(H + (long long)src * D) + c * 2;
    float4 v0 = hs[0];
    float4 v1 = hs[1];
    float* o = out + (long long)dst * D + c * 8;
    unsafeAtomicAdd(o + 0, v0.x * w);
    unsafeAtomicAdd(o + 1, v0.y * w);
    unsafeAtomicAdd(o + 2, v0.z * w);
    unsafeAtomicAdd(o + 3, v0.w * w);
    unsafeAtomicAdd(o + 4, v1.x * w);
    unsafeAtomicAdd(o + 5, v1.y * w);
    unsafeAtomicAdd(o + 6, v1.z * w);
    unsafeAtomicAdd(o + 7, v1.w * w);
}

// ---------------------------------------------------------------------------
extern "C" void kernel_launch(void* const* d_in, const int* in_sizes, int n_in,
                              void* d_out, int out_size, void* d_ws, size_t ws_size,
                              hipStream_t stream) {
    const float* x_dev = (const float*)d_in[0];
    const float* x_cat = (const float*)d_in[1];
    const float* x_app = (const float*)d_in[2];
    const int* ei_da = (const int*)d_in[3];
    const int* ei_ac = (const int*)d_in[4];
    const int* ei_aa = (const int*)d_in[5];
    const float* ew_da = (const float*)d_in[6];
    const float* ew_ac = (const float*)d_in[7];
    const float* ew_aa = (const float*)d_in[8];
    const float* W1d = (const float*)d_in[9],  *b1d = (const float*)d_in[10];
    const float* W1c = (const float*)d_in[11], *b1c = (const float*)d_in[12];
    const float* W1a = (const float*)d_in[13], *b1a = (const float*)d_in[14];
    const float* W2d = (const float*)d_in[15], *b2d = (const float*)d_in[16];
    const float* W2c = (const float*)d_in[17], *b2c = (const float*)d_in[18];
    const float* W2a = (const float*)d_in[19], *b2a = (const float*)d_in[20];

    const int Ndev = in_sizes[0] / D;
    const int Ncat = in_sizes[1] / D;
    const int Napp = in_sizes[2] / D;
    const int E_   = in_sizes[6];
    const long long TOT = (long long)Ndev + Ncat + Napp;

    float* h = (float*)d_ws;             // [TOT, D] transformed features
    float* y = h + TOT * D;              // [TOT, D] layer-1 aggregation
    float* h_dev = h;
    float* h_cat = h + (long long)Ndev * D;
    float* h_app = h + (long long)(Ndev + Ncat) * D;
    float* y_dev = y;
    float* y_cat = y + (long long)Ndev * D;
    float* y_app = y + (long long)(Ndev + Ncat) * D;

    auto gemm = [&](const float* A, const float* W, const float* b, float* Hout, int M) {
        int tiles_m = (M + 15) / 16;     // 4 waves per tile, 128 threads/block
        hgnn_gemm_bias_relu_wmma<<<tiles_m, 128, 0, stream>>>(A, W, b, Hout, M);
    };

    const int sblocks = (E_ * 32 + 255) / 256;

    // ---- Layer 1 ----
    gemm(x_dev, W1d, b1d, h_dev, Ndev);
    gemm(x_cat, W1c, b1c, h_cat, Ncat);
    gemm(x_app, W1a, b1a, h_app, Napp);

    int yn4 = (int)(TOT * D / 4);
    hgnn_zero_kernel<<<(yn4 + 255) / 256, 256, 0, stream>>>((float4*)y, yn4);

    hgnn_scatter_edges<<<sblocks, 256, 0, stream>>>(h_dev, ei_da, ew_da, y_app, E_);
    hgnn_scatter_edges<<<sblocks, 256, 0, stream>>>(h_app, ei_ac, ew_ac, y_cat, E_);
    hgnn_scatter_edges<<<sblocks, 256, 0, stream>>>(h_app, ei_aa, ew_aa, y_app, E_);

    // ---- Layer 2 ----
    gemm(y_dev, W2d, b2d, h_dev, Ndev);
    gemm(y_cat, W2c, b2c, h_cat, Ncat);
    gemm(y_app, W2a, b2a, h_app, Napp);

    float* od = (float*)d_out;           // (dev, cat, app) concatenated
    int on4 = out_size / 4;
    hgnn_zero_kernel<<<(on4 + 255) / 256, 256, 0, stream>>>((float4*)od, on4);

    float* o_cat = od + (long long)Ndev * D;
    float* o_app = od + (long long)(Ndev + Ncat) * D;
    hgnn_scatter_edges<<<sblocks, 256, 0, stream>>>(h_dev, ei_da, ew_da, o_app, E_);
    hgnn_scatter_edges<<<sblocks, 256, 0, stream>>>(h_app, ei_ac, ew_ac, o_cat, E_);
    hgnn_scatter_edges<<<sblocks, 256, 0, stream>>>(h_app, ei_aa, ew_aa, o_app, E_);
}